// Model_88553635709430
// MI455X (gfx1250) — compile-verified
//
#include <hip/hip_runtime.h>
#include <math.h>

typedef float v2f __attribute__((ext_vector_type(2)));
typedef float v8f __attribute__((ext_vector_type(8)));

#define N_DT 30000
#define N_OD 10000
#define D_DT 128
#define D_OD 64
#define HDIM 128
#define ODIM 64
#define E1N 500000
#define E2N 300000
#define ELN 200000

// ---------------------------------------------------------------- utilities
__global__ void k_zero(float* __restrict__ p, long n) {
    long i = (long)blockIdx.x * blockDim.x + threadIdx.x;
    if (i < n) p[i] = 0.0f;
}

// gather x[row[e]] (D=128) and atomic-add into msg[col[e]]; one wave per edge,
// each lane handles a float4 chunk. Lane with f4==0 also counts the edge.
__global__ void k_scatter128(const float* __restrict__ x,
                             const int* __restrict__ row,
                             const int* __restrict__ col,
                             float* __restrict__ msg,
                             float* __restrict__ cnt, int E) {
    long i = (long)blockIdx.x * blockDim.x + threadIdx.x;
    int e  = (int)(i >> 5);
    int f4 = ((int)i & 31) * 4;
    if (e >= E) return;
    int r = row[e], c = col[e];
    const float4 v = *(const float4*)(x + (long)r * 128 + f4);
    float* dst = msg + (long)c * 128 + f4;
    atomicAdd(dst + 0, v.x);
    atomicAdd(dst + 1, v.y);
    atomicAdd(dst + 2, v.z);
    atomicAdd(dst + 3, v.w);
    if (f4 == 0) atomicAdd(cnt + c, 1.0f);
}

__global__ void k_normalize(float* __restrict__ msg, const float* __restrict__ cnt,
                            long M, int D) {
    long i = (long)blockIdx.x * blockDim.x + threadIdx.x;
    if (i >= M * D) return;
    float c = cnt[i / D];
    msg[i] = msg[i] / fmaxf(c, 1.0f);
}

// ------------------------------------------------- fused dual-GEMM via WMMA
// C[M,N] = act( A1[M,K1] @ B1[K1,N]  +  A2[M,K2] @ B2[K2,N]  + bias[N] )
// One wave computes a 16x16 tile with V_WMMA_F32_16X16X4_F32, K step 4.
// Requires M % 16 == 0, N % 16 == 0, K % 4 == 0 (true for all call sites).
__global__ void k_gemm_wmma(const float* __restrict__ A1, const float* __restrict__ B1, int K1,
                            const float* __restrict__ A2, const float* __restrict__ B2, int K2,
                            const float* __restrict__ bias, float* __restrict__ C,
                            int M, int N, int doRelu) {
    int w    = threadIdx.x >> 5;
    int lane = threadIdx.x & 31;
    int half = lane >> 4;           // 0: K={0,1}, 1: K={2,3} of each step
    int l16  = lane & 15;
    int row0 = (blockIdx.y * 4 + w) * 16;
    int col0 = blockIdx.x * 16;
    if (row0 >= M) return;          // wave-uniform: EXEC stays all-ones below

    v8f acc = {};
    {
        const float* arow = A1 + (long)(row0 + l16) * K1;
        for (int k = 0; k < K1; k += 4) {
            int ka = k + 2 * half;
            v2f a = { arow[ka], arow[ka + 1] };
            v2f b = { B1[(long)ka * N + col0 + l16],
                      B1[(long)(ka + 1) * N + col0 + l16] };
            acc = __builtin_amdgcn_wmma_f32_16x16x4_f32(
                false, a, false, b, (short)0, acc, false, false);
        }
    }
    if (K2 > 0) {
        const float* arow = A2 + (long)(row0 + l16) * K2;
        for (int k = 0; k < K2; k += 4) {
            int ka = k + 2 * half;
            v2f a = { arow[ka], arow[ka + 1] };
            v2f b = { B2[(long)ka * N + col0 + l16],
                      B2[(long)(ka + 1) * N + col0 + l16] };
            acc = __builtin_amdgcn_wmma_f32_16x16x4_f32(
                false, a, false, b, (short)0, acc, false, false);
        }
    }
    float bv = bias[col0 + l16];
    for (int r = 0; r < 8; ++r) {
        int m  = row0 + r + 8 * half;      // C/D layout: VGPR r -> M=r (lo) / M=r+8 (hi)
        float v = acc[r] + bv;
        if (doRelu) v = fmaxf(v, 0.0f);
        C[(long)m * N + col0 + l16] = v;
    }
}

// --------------------------------------------------------- edge MLP decoder
// out[e] = sigmoid( relu( [z_od[src]|z_dt[dst]] @ W1 + b1 ) @ w2 + b2 )
// 16 edges per wave; A fragment gathered on the fly; 4 N-tiles of WMMA.
__global__ void k_decoder(const float* __restrict__ z_od, const float* __restrict__ z_dt,
                          const int* __restrict__ el_src, const int* __restrict__ el_dst,
                          const float* __restrict__ w1, const float* __restrict__ b1,
                          const float* __restrict__ w2, const float* __restrict__ b2,
                          float* __restrict__ out, int EL) {
    __shared__ float hid[4][16][64];
    int w    = threadIdx.x >> 5;
    int lane = threadIdx.x & 31;
    int half = lane >> 4;
    int l16  = lane & 15;
    int e0   = (blockIdx.x * 4 + w) * 16;

    if (e0 < EL) {                         // wave-uniform guard
        int src = el_src[e0 + l16];
        int dst = el_dst[e0 + l16];
        const float* zo = z_od + (long)src * 64;
        const float* zd = z_dt + (long)dst * 64;
        v8f acc0 = {}, acc1 = {}, acc2 = {}, acc3 = {};
        for (int k = 0; k < 128; k += 4) {
            int ka = k + 2 * half;         // ka even, so ka/ka+1 never straddle 64
            float ax = (ka < 64) ? zo[ka] : zd[ka - 64];
            float ay = (ka < 64) ? zo[ka + 1] : zd[ka + 1 - 64];
            v2f a = { ax, ay };
            v2f b0 = { w1[(long)ka * 64 +  0 + l16], w1[(long)(ka + 1) * 64 +  0 + l16] };
            v2f b1v= { w1[(long)ka * 64 + 16 + l16], w1[(long)(ka + 1) * 64 + 16 + l16] };
            v2f b2v= { w1[(long)ka * 64 + 32 + l16], w1[(long)(ka + 1) * 64 + 32 + l16] };
            v2f b3 = { w1[(long)ka * 64 + 48 + l16], w1[(long)(ka + 1) * 64 + 48 + l16] };
            acc0 = __builtin_amdgcn_wmma_f32_16x16x4_f32(false, a, false, b0, (short)0, acc0, false, false);
            acc1 = __builtin_amdgcn_wmma_f32_16x16x4_f32(false, a, false, b1v,(short)0, acc1, false, false);
            acc2 = __builtin_amdgcn_wmma_f32_16x16x4_f32(false, a, false, b2v,(short)0, acc2, false, false);
            acc3 = __builtin_amdgcn_wmma_f32_16x16x4_f32(false, a, false, b3, (short)0, acc3, false, false);
        }
        v8f accs[4] = { acc0, acc1, acc2, acc3 };
        for (int nt = 0; nt < 4; ++nt) {
            int n = nt * 16 + l16;
            float bb = b1[n];
            for (int r = 0; r < 8; ++r) {
                float v = accs[nt][r] + bb;
                hid[w][r + 8 * half][n] = fmaxf(v, 0.0f);
            }
        }
    }
    __syncthreads();
    if (e0 < EL && half == 0) {
        float s = 0.0f;
        for (int n = 0; n < 64; ++n) s += hid[w][l16][n] * w2[n];
        s += b2[0];
        out[e0 + l16] = 1.0f / (1.0f + __expf(-s));
    }
}

// ----------------------------------------------------------------- launcher
static inline long cdivl(long a, long b) { return (a + b - 1) / b; }

extern "C" void kernel_launch(void* const* d_in, const int* in_sizes, int n_in,
                              void* d_out, int out_size, void* d_ws, size_t ws_size,
                              hipStream_t stream) {
    const float* x_dt   = (const float*)d_in[0];
    const float* x_od   = (const float*)d_in[1];
    const int*   ei_dt  = (const int*)d_in[2];
    const int*   r1     = ei_dt;           // ei_dt[0]
    const int*   c1     = ei_dt + E1N;     // ei_dt[1]
    const int*   rev_src = (const int*)d_in[3];
    const int*   rev_dst = (const int*)d_in[4];
    const int*   el_src  = (const int*)d_in[5];
    const int*   el_dst  = (const int*)d_in[6];
    const float* od1_wl = (const float*)d_in[7];
    const float* od1_wr = (const float*)d_in[8];
    const float* od1_b  = (const float*)d_in[9];
    const float* od2_wl = (const float*)d_in[10];
    const float* od2_wr = (const float*)d_in[11];
    const float* od2_b  = (const float*)d_in[12];
    const float* od3_wl = (const float*)d_in[13];
    const float* od3_wr = (const float*)d_in[14];
    const float* od3_b  = (const float*)d_in[15];
    const float* od_lw  = (const float*)d_in[16];
    const float* od_lb  = (const float*)d_in[17];
    const float* dt1_wl = (const float*)d_in[18];
    const float* dt1_wr = (const float*)d_in[19];
    const float* dt1_b  = (const float*)d_in[20];
    const float* dt2_wl = (const float*)d_in[21];
    const float* dt2_wr = (const float*)d_in[22];
    const float* dt2_b  = (const float*)d_in[23];
    const float* dt_lw  = (const float*)d_in[24];
    const float* dt_lb  = (const float*)d_in[25];
    const float* dec_w1 = (const float*)d_in[26];
    const float* dec_b1 = (const float*)d_in[27];
    const float* dec_w2 = (const float*)d_in[28];
    const float* dec_b2 = (const float*)d_in[29];
    float* out = (float*)d_out;

    // workspace layout (floats) — ~72 MB total
    float* ws    = (float*)d_ws;
    float* buf0  = ws;                              // [N_DT x 128] agg_dt / agg_d
    float* buf1  = buf0 + (long)N_DT * HDIM;        // [N_DT x 128] h, later d2
    float* buf2  = buf1 + (long)N_DT * HDIM;        // [N_DT x 128] d1
    float* buf3  = buf2 + (long)N_DT * HDIM;        // [N_OD x 128] agg_h_od
    float* buf4  = buf3 + (long)N_OD * HDIM;        // [N_OD x 128] od2
    float* buf5  = buf4 + (long)N_OD * HDIM;        // [N_OD x 128] od3
    float* zod   = buf5 + (long)N_OD * HDIM;        // [N_OD x 64]
    float* zdt   = zod  + (long)N_OD * ODIM;        // [N_DT x 64]
    float* cntdt = zdt  + (long)N_DT * ODIM;        // [N_DT]
    float* cntod = cntdt + N_DT;                    // [N_OD]

    const dim3 B256(256), B128(128);

    // ---- 1) agg_dt = mean-agg of x_dt over ei_dt (shared by od1 and dt1)
    k_zero<<<cdivl((long)N_DT * HDIM, 256), B256, 0, stream>>>(buf0, (long)N_DT * HDIM);
    k_zero<<<cdivl(N_DT, 256), B256, 0, stream>>>(cntdt, N_DT);
    k_scatter128<<<cdivl((long)E1N * 32, 256), B256, 0, stream>>>(x_dt, r1, c1, buf0, cntdt, E1N);
    k_normalize<<<cdivl((long)N_DT * HDIM, 256), B256, 0, stream>>>(buf0, cntdt, N_DT, HDIM);

    // ---- 2) h = relu(agg_dt@od1_wl + x_dt@od1_wr + b);  d1 likewise for dt1
    {
        dim3 g(HDIM / 16, cdivl(N_DT, 64));
        k_gemm_wmma<<<g, B128, 0, stream>>>(buf0, od1_wl, D_DT, x_dt, od1_wr, D_DT,
                                            od1_b, buf1, N_DT, HDIM, 1);
        k_gemm_wmma<<<g, B128, 0, stream>>>(buf0, dt1_wl, D_DT, x_dt, dt1_wr, D_DT,
                                            dt1_b, buf2, N_DT, HDIM, 1);
    }

    // ---- 3) agg_h_od = mean-agg of h over reverse edges (shared by od2 and od3)
    k_zero<<<cdivl((long)N_OD * HDIM, 256), B256, 0, stream>>>(buf3, (long)N_OD * HDIM);
    k_zero<<<cdivl(N_OD, 256), B256, 0, stream>>>(cntod, N_OD);
    k_scatter128<<<cdivl((long)E2N * 32, 256), B256, 0, stream>>>(buf1, rev_src, rev_dst, buf3, cntod, E2N);
    k_normalize<<<cdivl((long)N_OD * HDIM, 256), B256, 0, stream>>>(buf3, cntod, N_OD, HDIM);

    // ---- 4) od2, od3, z_od
    {
        dim3 g(HDIM / 16, cdivl(N_OD, 64));
        k_gemm_wmma<<<g, B128, 0, stream>>>(buf3, od2_wl, HDIM, x_od, od2_wr, D_OD,
                                            od2_b, buf4, N_OD, HDIM, 1);
        k_gemm_wmma<<<g, B128, 0, stream>>>(buf3, od3_wl, HDIM, buf4, od3_wr, HDIM,
                                            od3_b, buf5, N_OD, HDIM, 1);
        dim3 gl(ODIM / 16, cdivl(N_OD, 64));
        k_gemm_wmma<<<gl, B128, 0, stream>>>(buf5, od_lw, HDIM, nullptr, nullptr, 0,
                                             od_lb, zod, N_OD, ODIM, 0);
    }

    // ---- 5) agg_d = mean-agg of d1 over ei_dt; d2; z_dt
    k_zero<<<cdivl((long)N_DT * HDIM, 256), B256, 0, stream>>>(buf0, (long)N_DT * HDIM);
    k_zero<<<cdivl(N_DT, 256), B256, 0, stream>>>(cntdt, N_DT);
    k_scatter128<<<cdivl((long)E1N * 32, 256), B256, 0, stream>>>(buf2, r1, c1, buf0, cntdt, E1N);
    k_normalize<<<cdivl((long)N_DT * HDIM, 256), B256, 0, stream>>>(buf0, cntdt, N_DT, HDIM);
    {
        dim3 g(HDIM / 16, cdivl(N_DT, 64));
        k_gemm_wmma<<<g, B128, 0, stream>>>(buf0, dt2_wl, HDIM, buf2, dt2_wr, HDIM,
                                            dt2_b, buf1, N_DT, HDIM, 1);
        dim3 gl(ODIM / 16, cdivl(N_DT, 64));
        k_gemm_wmma<<<gl, B128, 0, stream>>>(buf1, dt_lw, HDIM, nullptr, nullptr, 0,
                                             dt_lb, zdt, N_DT, ODIM, 0);
    }

    // ---- 6) edge decoder
    k_decoder<<<cdivl(ELN, 64), B128, 0, stream>>>(zod, zdt, el_src, el_dst,
                                                   dec_w1, dec_b1, dec_w2, dec_b2,
                                                   out, ELN);
}